// PilotAEQ_DP_80977313399027
// MI455X (gfx1250) — compile-verified
//
#include <hip/hip_runtime.h>

typedef float v2f __attribute__((ext_vector_type(2)));
typedef float v8f __attribute__((ext_vector_type(8)));

namespace {
constexpr int   kL       = 31;
constexpr int   kNumSamp = 200000;
constexpr int   kNPilot  = 100000;
constexpr int   kT       = 99984;
constexpr int   kB       = 16;
constexpr int   kNB      = kT / kB;     // 6249 blocks (exact)
constexpr int   kNT      = kT + 100;    // output columns per (ri,pol) plane
constexpr float kTwoLr   = 2.0e-3f;     // 2 * LR
constexpr int   kAStride = 132;         // padded LDS stride for A (bank-conflict free)
}

// Async copy of 16 bytes/lane from global memory into LDS (gfx1250).
// VDST operand carries the per-lane LDS byte address; completion tracked by ASYNCcnt.
__device__ __forceinline__ void async_b128(void* lds, const void* g) {
  unsigned int lo = (unsigned int)(unsigned long long)lds;   // low 32 bits = LDS offset
  unsigned long long ga = (unsigned long long)g;
  asm volatile("global_load_async_to_lds_b128 %0, %1, off"
               :: "v"(lo), "v"(ga) : "memory");
}

// ---------------------------------------------------------------------------
// Phase 1: per-block complex Gram matrices via V_WMMA_F32_16X16X4_F32.
// Block b covers symbols i = 16b + l (l=0..15). Gram rows r=0..15 are symbols
// j = 16b - 1 + r (row 0 = carried symbol from previous block).
//   G[r,l] = sum_{p,t} conj(wf_{j}) * wf_{i}
// Real decomposition over K=124 (= [re | im] x [2 pol x 31 taps]):
//   G_re = sum_k Ahat[r,k]*B1[k,l],  Ahat=[ar,ai], B1=[br,bi]
//   G_im = sum_k Ahat[r,k]*B2[k,l],  B2=[bi,-br]
// K = 124 = 31 chunks of 4 -> 62 WMMAs per block.
// ---------------------------------------------------------------------------
__global__ __launch_bounds__(32)
void aeq_gram_kernel(const float* __restrict__ yr, const float* __restrict__ yi,
                     float* __restrict__ ws) {
  __shared__ float ys[2][2][64];            // [comp][pol][sample-offset]
  __shared__ float Al[16 * kAStride];       // A[M][k], padded stride
  __shared__ float B1[124 * 16];            // B1[k][N]
  __shared__ float B2[124 * 16];            // B2[k][N]

  const int b   = blockIdx.x;
  const int tid = threadIdx.x;
  const long base = 2L * (kB * b) - 2;      // sample index of LDS offset 0

  // Stage 63 samples per (comp,pol). Offsets used: A -> 0..60, B -> 2..62.
  for (int idx = tid; idx < 2 * 2 * 63; idx += 32) {
    int comp = idx / 126; int rem = idx - comp * 126;
    int p = rem / 63;     int s = rem - p * 63;
    long g = base + s; if (g < 0) g = 0;    // block 0 row -1 is multiplied by e_prev=0
    const float* src = comp ? yi : yr;
    ys[comp][p][s] = src[(long)p * kNumSamp + g];
  }
  __syncthreads();

  // Build A (16 x 124): row M = symbol (16b-1+M), wf[t] = y[2j + 30 - t]
  for (int idx = tid; idx < 16 * 124; idx += 32) {
    int M = idx / 124; int k = idx - M * 124;
    int comp = (k >= 62); int kk = k - 62 * comp;
    int p = (kk >= 31);   int t = kk - 31 * p;
    Al[M * kAStride + k] = ys[comp][p][2 * M + 30 - t];
  }
  // Build B1/B2 (124 x 16): col N = symbol (16b + N)
  for (int idx = tid; idx < 124 * 16; idx += 32) {
    int k = idx >> 4; int N = idx & 15;
    int comp = (k >= 62); int kk = k - 62 * comp;
    int p = (kk >= 31);   int t = kk - 31 * p;
    int off = 2 * N + 32 - t;
    float re = ys[0][p][off];
    float im = ys[1][p][off];
    B1[idx] = comp ? im : re;
    B2[idx] = comp ? -re : im;
  }
  __syncthreads();

  // WMMA accumulation. A 16x4 frag: lanes 0-15 hold K=k0,k0+1; lanes 16-31 K=k0+2,k0+3.
  // B 4x16 frag mirrors: VGPR0 rows k0 / k0+2, VGPR1 rows k0+1 / k0+3, col = lane&15.
  const int Ml = tid & 15;
  const int kh = (tid >> 4) * 2;
  v8f accRe = {0.f,0.f,0.f,0.f,0.f,0.f,0.f,0.f};
  v8f accIm = {0.f,0.f,0.f,0.f,0.f,0.f,0.f,0.f};
  for (int c = 0; c < 31; ++c) {
    int k0 = 4 * c + kh;
    v2f a, b1, b2;
    a.x  = Al[Ml * kAStride + k0];
    a.y  = Al[Ml * kAStride + k0 + 1];
    b1.x = B1[k0 * 16 + Ml];
    b1.y = B1[(k0 + 1) * 16 + Ml];
    b2.x = B2[k0 * 16 + Ml];
    b2.y = B2[(k0 + 1) * 16 + Ml];
    accRe = __builtin_amdgcn_wmma_f32_16x16x4_f32(false, a, false, b1,
                                                  (short)0, accRe, false, false);
    accIm = __builtin_amdgcn_wmma_f32_16x16x4_f32(false, a, false, b2,
                                                  (short)0, accIm, false, false);
  }

  // D 16x16 f32 layout: VGPR v -> row v (lanes 0-15) / row v+8 (lanes 16-31), col = lane&15.
  float* G = ws + (size_t)b * 512;
  const int rbase = (tid < 16) ? 0 : 8;
  const int l = tid & 15;
  for (int v = 0; v < 8; ++v) {
    G[(rbase + v) * 16 + l]       = accRe[v];
    G[256 + (rbase + v) * 16 + l] = accIm[v];
  }
}

// ---------------------------------------------------------------------------
// Phase 2: sequential block recurrence (single wave), double-buffered with
// GLOBAL_LOAD_ASYNC_TO_LDS_B128 prefetch of the next block's window + Gram.
// Exact LMS semantics:
//   out_{s0+l} = <taps_{s0}, wf_l> + 2mu*( e_prev*G[0,l]*[l>=1]
//                 + sum_{m<=l-2} e_m*G[m+1,l] ),  e = pilot - out
//   taps_{s0+16} = taps_{s0} + 2mu*( u_{s0-1} + sum_{j=s0..s0+14} u_j )
// ---------------------------------------------------------------------------
__global__ __launch_bounds__(32)
void aeq_seq_kernel(const float* __restrict__ yr, const float* __restrict__ yi,
                    const float* __restrict__ pr, const float* __restrict__ pi,
                    const float* __restrict__ tr0, const float* __restrict__ ti0,
                    const float* __restrict__ ws, float* __restrict__ out) {
  __shared__ float tapsR[4][32], tapsI[4][32];
  __shared__ float ys[2][2][2][64];         // [buf][comp][pol][sample-offset]
  __shared__ float Gbuf[2][512];            // [buf][ re(256) | im(256) ]
  __shared__ float eRs[2][16], eIs[2][16];
  __shared__ float epR[2], epI[2];

  const int tid = threadIdx.x;
  const int l   = tid & 15;
  const int pol = tid >> 4;

  // Zero-fill the leading 100 columns of each (ri,pol) plane.
  for (int idx = tid; idx < 400; idx += 32) {
    int plane = idx / 100; int t = idx - plane * 100;
    out[(size_t)plane * kNT + t] = 0.0f;
  }
  // Load initial taps; zero carried error.
  for (int idx = tid; idx < 124; idx += 32) {
    int row = idx / 31; int t = idx - row * 31;
    tapsR[row][t] = tr0[row * 31 + t];
    tapsI[row][t] = ti0[row * 31 + t];
  }
  if (tid < 2) { epR[tid] = 0.0f; epI[tid] = 0.0f; }

  // Prologue: block 0 loaded synchronously (its window starts 2 samples before
  // the array; clamp -- those entries are multiplied by e_prev = 0).
  for (int idx = tid; idx < 2 * 2 * 63; idx += 32) {
    int comp = idx / 126; int rem = idx - comp * 126;
    int p = rem / 63;     int s = rem - p * 63;
    long g = -2L + s; if (g < 0) g = 0;
    ys[0][comp][p][s] = (comp ? yi : yr)[(long)p * kNumSamp + g];
  }
  for (int idx = tid; idx < 512; idx += 32) Gbuf[0][idx] = ws[idx];
  __syncthreads();

  int cur = 0;
  for (int b = 0; b < kNB; ++b) {
    const int nxt = cur ^ 1;
    // Prefetch next block (base >= 30 for b+1 >= 1, no clamp needed):
    // 2 instrs for the sample window (16 lanes x 16B per pol), 4 for the Gram.
    if (b + 1 < kNB) {
      const long basen = 2L * kB * (b + 1) - 2;
      const int p  = tid >> 4;
      const int j4 = (tid & 15) * 4;
      async_b128(&ys[nxt][0][p][j4], yr + (long)p * kNumSamp + basen + j4);
      async_b128(&ys[nxt][1][p][j4], yi + (long)p * kNumSamp + basen + j4);
      const float* Gn = ws + (size_t)(b + 1) * 512;
      const int q4 = tid * 4;
      async_b128(&Gbuf[nxt][q4],       Gn + q4);
      async_b128(&Gbuf[nxt][128 + q4], Gn + 128 + q4);
      async_b128(&Gbuf[nxt][256 + q4], Gn + 256 + q4);
      async_b128(&Gbuf[nxt][384 + q4], Gn + 384 + q4);
      // <=6 outstanding == the 6 just issued; current buffer is complete
      // (async loads retire in order).
      asm volatile("s_wait_asynccnt 0x6" ::: "memory");
    } else {
      asm volatile("s_wait_asynccnt 0x0" ::: "memory");
    }

    const int s0 = kB * b;

    // Base projection with block-start taps.
    // pol0: taps0*wf0 + taps1*wf1 ; pol1: taps2*wf1 + taps3*wf0
    const int rowA = pol ? 2 : 0; const int pA = pol ? 1 : 0;
    const int rowB = pol ? 3 : 1; const int pB = pol ? 0 : 1;
    float bR = 0.0f, bI = 0.0f;
    for (int t = 0; t < kL; ++t) {
      int off = 2 * l + 32 - t;
      float wr = ys[cur][0][pA][off], wi = ys[cur][1][pA][off];
      float cr = tapsR[rowA][t],      ci = tapsI[rowA][t];
      bR += cr * wr - ci * wi;  bI += cr * wi + ci * wr;
      wr = ys[cur][0][pB][off]; wi = ys[cur][1][pB][off];
      cr = tapsR[rowB][t];      ci = tapsI[rowB][t];
      bR += cr * wr - ci * wi;  bI += cr * wi + ci * wr;
    }
    // Carried-error term (u from last symbol of previous block).
    const float er0 = epR[pol], ei0 = epI[pol];
    if (l >= 1) {
      float gr = Gbuf[cur][l], gi = Gbuf[cur][256 + l];   // G[0, l]
      bR += kTwoLr * (er0 * gr - ei0 * gi);
      bI += kTwoLr * (er0 * gi + ei0 * gr);
    }
    const int pidx = pol * kNPilot + 7 + s0 + l;          // EQ_OFFSET = 7
    const float pRv = pr[pidx], pIv = pi[pidx];

    // 16-step wavefront recurrence: each new e_m broadcast via shuffle.
    float accR = bR, accI = bI;
    float myER = 0.0f, myEI = 0.0f;
    for (int m = 0; m < 16; ++m) {
      float emRs = 0.0f, emIs = 0.0f;
      if (l == m) {
        myER = pRv - accR; myEI = pIv - accI;
        emRs = myER;       emIs = myEI;
      }
      int src = (tid & 16) + m;                           // same-pol source lane
      float emR = __shfl(emRs, src, 32);
      float emI = __shfl(emIs, src, 32);
      if (l >= m + 2) {
        float gr = Gbuf[cur][(m + 1) * 16 + l];
        float gi = Gbuf[cur][256 + (m + 1) * 16 + l];
        accR += kTwoLr * (emR * gr - emI * gi);
        accI += kTwoLr * (emR * gi + emI * gr);
      }
    }
    eRs[pol][l] = myER; eIs[pol][l] = myEI;

    // out = pilot - e ; planes: [ri][pol][col]
    const size_t col = (size_t)(100 + s0 + l);
    out[(size_t)pol * kNT + col]       = pRv - myER;      // real plane
    out[(size_t)(2 + pol) * kNT + col] = pIv - myEI;      // imag plane
    __syncthreads();

    // Rank-16 tap update: u rows {e0*conj(w0), e0*conj(w1), e1*conj(w1), e1*conj(w0)}
    for (int idx = tid; idx < 124; idx += 32) {
      int row = idx / 31; int t = idx - row * 31;
      int p  = (row == 1 || row == 2) ? 1 : 0;
      int es = row >> 1;
      float sR = 0.0f, sI = 0.0f;
      {   // r = 0: carried error, window of symbol s0-1
        int off = 30 - t;
        float wr = ys[cur][0][p][off], wi = ys[cur][1][p][off];
        float cr = epR[es], ci = epI[es];
        sR += cr * wr + ci * wi;      // e * conj(w)
        sI += ci * wr - cr * wi;
      }
      for (int r = 1; r <= 15; ++r) { // e_0 .. e_14 (e_15 carried forward)
        int off = 2 * r + 30 - t;
        float wr = ys[cur][0][p][off], wi = ys[cur][1][p][off];
        float cr = eRs[es][r - 1], ci = eIs[es][r - 1];
        sR += cr * wr + ci * wi;
        sI += ci * wr - cr * wi;
      }
      tapsR[row][t] += kTwoLr * sR;
      tapsI[row][t] += kTwoLr * sI;
    }
    __syncthreads();
    if (tid < 2) { epR[tid] = eRs[tid][15]; epI[tid] = eIs[tid][15]; }
    __syncthreads();
    cur = nxt;
  }
}

extern "C" void kernel_launch(void* const* d_in, const int* in_sizes, int n_in,
                              void* d_out, int out_size, void* d_ws, size_t ws_size,
                              hipStream_t stream) {
  (void)in_sizes; (void)n_in; (void)out_size; (void)ws_size;
  const float* yr  = (const float*)d_in[0];   // (2, 200000)
  const float* yi  = (const float*)d_in[1];
  const float* pr  = (const float*)d_in[2];   // (2, 100000)
  const float* pi  = (const float*)d_in[3];
  const float* tr0 = (const float*)d_in[4];   // (4, 31)
  const float* ti0 = (const float*)d_in[5];
  float* ws  = (float*)d_ws;                  // needs kNB*512*4 = ~12.8 MB
  float* out = (float*)d_out;                 // (2, 2, T+100)

  aeq_gram_kernel<<<kNB, 32, 0, stream>>>(yr, yi, ws);
  aeq_seq_kernel<<<1, 32, 0, stream>>>(yr, yi, pr, pi, tr0, ti0, ws, out);
}